// AsymmetricEMA_19490561590050
// MI455X (gfx1250) — compile-verified
//
#include <hip/hip_runtime.h>

#define AS1 __attribute__((address_space(1)))
#define AS3 __attribute__((address_space(3)))

typedef int v4i __attribute__((vector_size(4 * sizeof(int))));

#ifndef __has_builtin
#define __has_builtin(x) 0
#endif

#if __has_builtin(__builtin_amdgcn_global_load_async_to_lds_b128)
#define HAVE_ASYNC_LDS 1
#else
#define HAVE_ASYNC_LDS 0
#endif

// Problem shape (fixed by the reference)
constexpr int B = 16;
constexpr int T = 4096;
constexpr int C = 1024;

// Tiling
constexpr int CH   = 64;                    // channels per block (== threads per block, 2 waves)
constexpr int TT   = 32;                    // timesteps per LDS tile
constexpr int NBUF = 8;                     // ring depth
constexpr int TILE_BYTES = TT * CH * 4;     // 8192 B per tile
constexpr int ISSUES = TILE_BYTES / (CH * 16); // 8 async_b128 issues per tile per wave
constexpr int NTILES = T / TT;              // 128
static_assert((NBUF - 1) * ISSUES <= 62, "ASYNCcnt wait threshold must fit 6-bit counter");

// EMA constants
constexpr float A_RISE = 0.5f;
constexpr float A_FALL = 0.99f;

#if HAVE_ASYNC_LDS

__device__ __forceinline__ void wait_tiles_resident() {
#if __has_builtin(__builtin_amdgcn_s_wait_asynccnt)
    __builtin_amdgcn_s_wait_asynccnt((NBUF - 1) * ISSUES);
#else
    asm volatile("s_wait_asynccnt 56" ::: "memory");
#endif
}

// Copy one [TT x CH] f32 tile x[b, t0:t0+TT, c0:c0+CH] into LDS ring slot `buf`.
// Each lane moves 16 contiguous bytes per issue; row0 = tid>>4 (4 rows per issue
// across 64 threads), global row stride = C*4 bytes, LDS row stride = CH*4 bytes.
__device__ __forceinline__ void issue_tile(const float* __restrict__ x,
                                           AS3 char* lds0,
                                           int b, int c0, int t0, int buf, int tid) {
    const int row0 = tid >> 4;
    const int colb = (tid & 15) << 4;
    const char* gp = (const char*)x
                   + (((size_t)b * T + (size_t)t0) * C + (size_t)c0) * sizeof(float)
                   + (size_t)row0 * (C * sizeof(float)) + (size_t)colb;
    AS3 char* lp = lds0 + (size_t)buf * TILE_BYTES + row0 * (CH * 4) + colb;
#pragma unroll
    for (int i = 0; i < ISSUES; ++i) {
        __builtin_amdgcn_global_load_async_to_lds_b128(
            (AS1 v4i*)(char*)gp, (AS3 v4i*)lp, /*offset=*/0, /*cpol=*/0);
        gp += 4 * C * sizeof(float);   // 4 rows forward in global
        lp += 4 * CH * sizeof(float);  // 4 rows forward in LDS
    }
}

// y = alpha*y_prev + (1-alpha)*x  ==  x + alpha*(y_prev - x)
// 4-deep dependency chain: sub -> cmp -> cndmask -> fma
__device__ __forceinline__ float ema_step(float yv, float xv) {
    const float d = yv - xv;
    const float a = (d > 0.0f) ? A_FALL : A_RISE;
    return __builtin_fmaf(a, d, xv);
}

__global__ __launch_bounds__(CH) void ema_async_kernel(const float* __restrict__ x,
                                                       float* __restrict__ y) {
    __shared__ __align__(16) char smem[NBUF * TILE_BYTES];  // 64 KB ring
    AS3 char* lds0 = (AS3 char*)smem;

    const int tid = threadIdx.x;
    const int blk = blockIdx.x;
    const int b   = blk >> 4;          // C/CH == 16 channel groups per batch
    const int c0  = (blk & 15) * CH;
    const int ch  = c0 + tid;

    // Prologue: fill the ring (hardware throttles ASYNCcnt at 63; harmless stall)
#pragma unroll
    for (int p = 0; p < NBUF; ++p)
        issue_tile(x, lds0, b, c0, p * TT, p, tid);

    float yv = 0.0f;
    const size_t out0 = (size_t)b * T * C + (size_t)ch;  // index of (b, t=0, ch)

    // ---- tile 0: first frame passes x through ----
    {
        wait_tiles_resident();
        __syncthreads();
        AS3 const float* lbuf = (AS3 const float*)lds0;
        yv = lbuf[tid];                                  // t = 0
        __builtin_nontemporal_store(yv, &y[out0]);
#pragma unroll
        for (int tt = 1; tt < TT; ++tt) {
            const float xv = lbuf[tt * CH + tid];
            yv = ema_step(yv, xv);
            __builtin_nontemporal_store(yv, &y[out0 + (size_t)tt * C]);
        }
        __syncthreads();                                 // all waves done reading buf 0
        issue_tile(x, lds0, b, c0, NBUF * TT, 0, tid);   // refill buf 0 with tile 8
    }

    // ---- steady state ----
    for (int tile = 1; tile < NTILES; ++tile) {
        wait_tiles_resident();
        __syncthreads();
        const int buf = tile & (NBUF - 1);
        AS3 const float* lbuf = (AS3 const float*)(lds0 + (size_t)buf * TILE_BYTES);
        const size_t oidx = out0 + (size_t)tile * TT * C;
#pragma unroll
        for (int tt = 0; tt < TT; ++tt) {
            const float xv = lbuf[tt * CH + tid];
            yv = ema_step(yv, xv);
            __builtin_nontemporal_store(yv, &y[oidx + (size_t)tt * C]);
        }
        __syncthreads();
        const int nt = tile + NBUF;
        if (nt < NTILES)
            issue_tile(x, lds0, b, c0, nt * TT, buf, tid);
    }
}

#else  // !HAVE_ASYNC_LDS — direct-load fallback (correct, fewer CDNA5 features)

__global__ __launch_bounds__(256) void ema_simple_kernel(const float* __restrict__ x,
                                                         float* __restrict__ y) {
    const int gid = blockIdx.x * blockDim.x + threadIdx.x;  // 0..16383
    const int b   = gid >> 10;
    const int ch  = gid & (C - 1);
    size_t idx = (size_t)b * T * C + (size_t)ch;

    float yv = x[idx];
    __builtin_nontemporal_store(yv, &y[idx]);
    idx += C;
#pragma unroll 8
    for (int t = 1; t < T; ++t, idx += C) {
        __builtin_prefetch(&x[idx + 16 * C], 0, 0);  // global_prefetch_b8, ~16 rows ahead
        const float xv = x[idx];
        const float d  = yv - xv;
        const float a  = (d > 0.0f) ? A_FALL : A_RISE;
        yv = __builtin_fmaf(a, d, xv);
        __builtin_nontemporal_store(yv, &y[idx]);
    }
}

#endif

extern "C" void kernel_launch(void* const* d_in, const int* in_sizes, int n_in,
                              void* d_out, int out_size, void* d_ws, size_t ws_size,
                              hipStream_t stream) {
    const float* x = (const float*)d_in[0];
    float* y = (float*)d_out;
    (void)in_sizes; (void)n_in; (void)out_size; (void)d_ws; (void)ws_size;

#if HAVE_ASYNC_LDS
    // 256 blocks (16 batches x 16 channel-groups) x 64 threads (2 waves)
    ema_async_kernel<<<dim3(B * (C / CH)), dim3(CH), 0, stream>>>(x, y);
#else
    ema_simple_kernel<<<dim3((B * C) / 256), dim3(256), 0, stream>>>(x, y);
#endif
}